// BasicBlock_6665789244013
// MI455X (gfx1250) — compile-verified
//
#include <hip/hip_runtime.h>
#include <hip/hip_bf16.h>
#include <math.h>

// ---------------------------------------------------------------------------
// Shapes (from reference): N=8, L=512, E=128, H_DIM=256, Q_DIM=512,
// IN_DIM=512, N_HEADS=4, N_TYPE=3, FH=64, LSTM_IN=512, LSTM_H=128.
// Outputs: doc_out (8,512,256) | entity_state2 (8,128,256) | softmask (8,128,1)
// ---------------------------------------------------------------------------

typedef float        v2f __attribute__((ext_vector_type(2)));
typedef float        v8f __attribute__((ext_vector_type(8)));
typedef unsigned int v4u __attribute__((ext_vector_type(4)));
typedef int          v8i __attribute__((ext_vector_type(8)));
typedef int          v4i __attribute__((ext_vector_type(4)));

#define LRELU 0.2f

static __device__ __forceinline__ float sigf(float x) {
  return 1.0f / (1.0f + __expf(-x));
}

// ---------------------------------------------------------------------------
// fp32 WMMA tile loops: one 16x16 C tile per wave32, K stepped by 4 with
// V_WMMA_F32_16X16X4_F32.  Per ISA VGPR layouts:
//   A (16x4): lane r (0..15) holds A[r][2*hf+0], A[r][2*hf+1]   (hf = lane>>4)
//   B (4x16): lane r holds B[2*hf+0][r], B[2*hf+1][r]
//   C/D     : acc[v] = D[v + 8*hf][r]
// ---------------------------------------------------------------------------

// A row-major (lda), B row-major KxN (ldb)
static __device__ __forceinline__ v8f wmma_rr(const float* __restrict__ A, int lda,
                                              const float* __restrict__ B, int ldb,
                                              int K, int r, int hf, v8f acc) {
  for (int k = 0; k < K; k += 4) {
    const int ka = k + 2 * hf;
    v2f a; a.x = A[r * lda + ka];      a.y = A[r * lda + ka + 1];
    v2f b; b.x = B[ka * ldb + r];      b.y = B[(ka + 1) * ldb + r];
    acc = __builtin_amdgcn_wmma_f32_16x16x4_f32(false, a, false, b,
                                                (short)0, acc, false, false);
  }
  return acc;
}

// A column-strided: element (m,k) at A[k*astr + m] (A pre-offset by m0), B row-major
static __device__ __forceinline__ v8f wmma_cr(const float* __restrict__ A, int astr,
                                              const float* __restrict__ B, int ldb,
                                              int K, int r, int hf, v8f acc) {
  for (int k = 0; k < K; k += 4) {
    const int ka = k + 2 * hf;
    v2f a; a.x = A[ka * astr + r];      a.y = A[(ka + 1) * astr + r];
    v2f b; b.x = B[ka * ldb + r];       b.y = B[(ka + 1) * ldb + r];
    acc = __builtin_amdgcn_wmma_f32_16x16x4_f32(false, a, false, b,
                                                (short)0, acc, false, false);
  }
  return acc;
}

// ---------------------------------------------------------------------------
// 1) Entity pooling: max/mean over L per (n,e,d).  Exploits ~3% sparsity of
//    ent_mapping: stage the mask row in LDS, skip doc loads where zero.
// ---------------------------------------------------------------------------
__global__ void k_entity_pool(const float* __restrict__ doc,
                              const float* __restrict__ emap,
                              float* __restrict__ ent) {
  const int row = blockIdx.x;          // n*128 + e
  const int n = row >> 7;
  const int d = threadIdx.x;           // 0..255
  __shared__ float ms[512];
  ms[d]       = emap[row * 512 + d];
  ms[d + 256] = emap[row * 512 + d + 256];
  __syncthreads();
  float mx = -3.4e38f, sum = 0.f, len = 0.f;
  bool has_zero = false;
  const float* dn = doc + n * (512 * 256) + d;
  for (int l = 0; l < 512; ++l) {
    const float m = ms[l];
    len += m;
    if (m != 0.f) {
      const float v = m * dn[l * 256];
      mx = fmaxf(mx, v);
      sum += v;
    } else {
      has_zero = true;
    }
  }
  if (has_zero) mx = fmaxf(mx, 0.f);
  if (len == 0.f) len = 1.f;
  ent[row * 512 + d]       = mx;
  ent[row * 512 + 256 + d] = sum / len;
}

// 2) query = query_vec @ query_weight  (8x512 @ 512x512)
__global__ void k_query_proj(const float* __restrict__ qv,
                             const float* __restrict__ qw,
                             float* __restrict__ query) {
  const int id = blockIdx.x * blockDim.x + threadIdx.x;  // 0..4095
  const int n = id >> 9, j = id & 511;
  float s = 0.f;
  for (int k = 0; k < 512; ++k) s += qv[n * 512 + k] * qw[k * 512 + j];
  query[id] = s;
}

// 3) softmask[n,e] = dot(ent_state, query)/512 * ent_mask ; adj_mask = sigmoid
__global__ void k_softmask(const float* __restrict__ ent,
                           const float* __restrict__ query,
                           const float* __restrict__ emask,
                           float* __restrict__ out3,
                           float* __restrict__ adjm) {
  const int id = blockIdx.x * blockDim.x + threadIdx.x;  // 0..1023 (n*128+e)
  const int n = id >> 7;
  float s = 0.f;
  for (int k = 0; k < 512; ++k) s += ent[id * 512 + k] * query[n * 512 + k];
  const float sm = (s * (1.0f / 512.0f)) * emask[id];    // temp = sqrt(512*512)
  out3[id] = sm;
  adjm[id] = sigf(sm);
}

// 4) h_all[ht][m][f] = ent(1024x512) @ W_type[ht](512x64)   [WMMA]
__global__ void k_hall(const float* __restrict__ ent,
                       const float* __restrict__ Wt,
                       float* __restrict__ hall) {
  const int wid  = (blockIdx.x * blockDim.x + threadIdx.x) >> 5;  // 0..3071
  const int lane = threadIdx.x & 31, hf = lane >> 4, r = lane & 15;
  const int nt = wid & 3, mt = (wid >> 2) & 63, ht = wid >> 8;
  v8f acc = {};
  acc = wmma_rr(ent + mt * 16 * 512, 512,
                Wt + ht * 32768 + nt * 16, 64, 512, r, hf, acc);
  float* D = hall + ht * 65536 + (mt * 16) * 64 + nt * 16;
#pragma unroll
  for (int v = 0; v < 8; ++v) D[(v + 8 * hf) * 64 + r] = acc[v];
}

// 5a) qg1 = relu(query_vec @ qattn_W1[ht])   (12,8,128)
__global__ void k_qg1(const float* __restrict__ qv,
                      const float* __restrict__ W1,
                      float* __restrict__ qg1) {
  const int id = blockIdx.x * blockDim.x + threadIdx.x;  // 0..12287
  const int ht = id >> 10, rem = id & 1023, n = rem >> 7, g = rem & 127;
  float s = 0.f;
  for (int k = 0; k < 512; ++k)
    s += qv[n * 512 + k] * W1[ht * 65536 + k * 128 + g];
  qg1[id] = fmaxf(s, 0.f);
}

// 5b) qg2 = sigmoid(qg1 @ qattn_W2[ht])   (12,8,128)
__global__ void k_qg2(const float* __restrict__ qg1,
                      const float* __restrict__ W2,
                      float* __restrict__ qg2) {
  const int id = blockIdx.x * blockDim.x + threadIdx.x;
  const int ht = id >> 10, rem = id & 1023, n = rem >> 7, g = rem & 127;
  float s = 0.f;
  for (int k = 0; k < 128; ++k)
    s += qg1[ht * 1024 + n * 128 + k] * W2[ht * 16384 + k * 128 + g];
  qg2[id] = sigf(s);
}

// 6) src/dst[ht][n][e] = sum_f h_all*g*a
__global__ void k_srcdst(const float* __restrict__ hall,
                         const float* __restrict__ qg2,
                         const float* __restrict__ at,
                         float* __restrict__ src,
                         float* __restrict__ dst) {
  const int id = blockIdx.x * blockDim.x + threadIdx.x;  // 0..12287
  const int ht = id >> 10, rem = id & 1023, n = rem >> 7;
  float s = 0.f, d = 0.f;
  const float* h = hall + ht * 65536 + rem * 64;
  const float* g = qg2 + ht * 1024 + n * 128;
  const float* a = at + ht * 128;
  for (int f = 0; f < 64; ++f) {
    const float hv = h[f];
    s += hv * g[f] * a[f];
    d += hv * g[64 + f] * a[64 + f];
  }
  src[id] = s;
  dst[id] = d;
}

// 7) h_last[h][n][e][f] = h_all[h][2][n][e][f] * adj_mask[n][e]
__global__ void k_hlast(const float* __restrict__ hall,
                        const float* __restrict__ adjm,
                        float* __restrict__ hlast) {
  const int id = blockIdx.x * blockDim.x + threadIdx.x;  // 0..262143
  const int h = id >> 16, rem = id & 65535, ne = rem >> 6;
  hlast[id] = hall[(h * 3 + 2) * 65536 + rem] * adjm[ne];
}

// 8) coefs = softmax over j of masked leaky-relu(src_i + dst_j) per type match
__global__ void k_coefs(const float* __restrict__ src,
                        const float* __restrict__ dst,
                        const int* __restrict__ adj,
                        float* __restrict__ coefs) {
  const int bid = blockIdx.x;                // 0..4095 = (h,n,i)
  const int h = bid >> 10, n = (bid >> 7) & 7, i = bid & 127;
  const int j = threadIdx.x;                 // 0..127
  __shared__ float red[128];
  const int a = adj[n * 16384 + i * 128 + j];
  float s;
  if (a > 0) {
    const int t = a - 1;
    float v = src[(h * 3 + t) * 1024 + n * 128 + i] +
              dst[(h * 3 + t) * 1024 + n * 128 + j];
    s = (v > 0.f) ? v : LRELU * v;
  } else {
    s = -1e30f;
  }
  red[j] = s; __syncthreads();
  for (int off = 64; off > 0; off >>= 1) {
    if (j < off) red[j] = fmaxf(red[j], red[j + off]);
    __syncthreads();
  }
  const float mx = red[0]; __syncthreads();
  const float e = __expf(s - mx);
  red[j] = e; __syncthreads();
  for (int off = 64; off > 0; off >>= 1) {
    if (j < off) red[j] += red[j + off];
    __syncthreads();
  }
  coefs[(h * 8 + n) * 16384 + i * 128 + j] = e / red[0];
}

// 9) heads = coefs(128x128) @ h_last(128x64) per (h,n); relu -> entity_state2
__global__ void k_heads(const float* __restrict__ coefs,
                        const float* __restrict__ hlast,
                        float* __restrict__ out2) {
  const int wid  = (blockIdx.x * blockDim.x + threadIdx.x) >> 5;  // 0..1023
  const int lane = threadIdx.x & 31, hf = lane >> 4, r = lane & 15;
  const int nt = wid & 3, mt = (wid >> 2) & 7, hn = wid >> 5;
  const int h = hn >> 3, n = hn & 7;
  v8f acc = {};
  acc = wmma_rr(coefs + hn * 16384 + mt * 16 * 128, 128,
                hlast + hn * 8192 + nt * 16, 64, 128, r, hf, acc);
  float* D = out2 + n * 32768 + h * 64 + nt * 16;
#pragma unroll
  for (int v = 0; v < 8; ++v)
    D[(mt * 16 + v + 8 * hf) * 256 + r] = fmaxf(acc[v], 0.f);
}

// 10) expand[n][l][d] = sum_e ent2[n][e][d] * map[n][e][l] -> inp[:, :256]
__global__ void k_expand(const float* __restrict__ emap,
                         const float* __restrict__ ent2,
                         float* __restrict__ inp) {
  const int wid  = (blockIdx.x * blockDim.x + threadIdx.x) >> 5;  // 0..4095
  const int lane = threadIdx.x & 31, hf = lane >> 4, r = lane & 15;
  const int nt = wid & 15, mt = (wid >> 4) & 31, n = wid >> 9;
  v8f acc = {};
  // A[m=l][k=e] = map[n][e][l]  (column-strided, stride L=512)
  acc = wmma_cr(emap + n * 65536 + mt * 16, 512,
                ent2 + n * 32768 + nt * 16, 256, 128, r, hf, acc);
  float* D = inp + n * 262144 + (mt * 16) * 512 + nt * 16;
#pragma unroll
  for (int v = 0; v < 8; ++v) D[(v + 8 * hf) * 512 + r] = acc[v];
}

// 11) inp[:, 256:512] = doc_state
__global__ void k_copy_doc(const float* __restrict__ doc,
                           float* __restrict__ inp) {
  const int id = blockIdx.x * blockDim.x + threadIdx.x;  // 0..1048575
  const int n = id >> 17, rem = id & 131071, l = rem >> 8, d = rem & 255;
  inp[n * 262144 + l * 512 + 256 + d] = doc[id];
}

// ---------------------------------------------------------------------------
// 12) pre = inp(4096x512) @ Wih^T + bih + bhh   [WMMA + TDM]
// Each block owns one N-tile (16 gate rows of Wih = 16x512 f32 = 32KB) and
// stages it into LDS with a single Tensor Data Mover descriptor (2D tile,
// data_size=4B, tile=512x16, stride=512), issued by wave 0 and synchronized
// with s_wait_tensorcnt + workgroup barrier.  The block's 4 waves then each
// compute one 16x16 output tile, reading B fragments from LDS (ds_load_b64)
// and A fragments streamed from global (global_load_b64).
// ---------------------------------------------------------------------------
__global__ __launch_bounds__(128) void k_pre(const float* __restrict__ inp,
                                             const float* __restrict__ Wih,
                                             const float* __restrict__ bih,
                                             const float* __restrict__ bhh,
                                             float* __restrict__ pre) {
  __shared__ float ldsB[16 * 512];                 // B panel: BT[n][k], ld 512
  const int nt   = blockIdx.x & 31;                // N-tile (gate-row group)
  const int mtg  = blockIdx.x >> 5;                // 0..63 (group of 4 M-tiles)
  const int wave = threadIdx.x >> 5;
  const int lane = threadIdx.x & 31, hf = lane >> 4, r = lane & 15;

  if (threadIdx.x < 32) {
    // Tensor DMA descriptor (D#): groups per CDNA5 ISA
    const unsigned long long ga =
        (unsigned long long)(const void*)(Wih + nt * 16 * 512);
    const unsigned lb = (unsigned)(unsigned long long)(void*)ldsB;
    v4u g0;
    g0.x = 1u;                                        // count=1 (valid), 2D
    g0.y = lb;                                        // lds_addr
    g0.z = (unsigned)(ga & 0xFFFFFFFFu);              // global_addr[31:0]
    g0.w = (unsigned)((ga >> 32) & 0x1FFFFFFu)        // global_addr[56:32]
           | (2u << 30);                              // type=2 (image)
    v8i g1;
    g1[0] = 0x20000;            // wg_mask=0 | data_size=2 (4B)
    g1[1] = (int)(512u << 16);  // tensor_dim0[15:0]=512 (in high half)
    g1[2] = (int)(16u << 16);   // tensor_dim0 hi=0 | tensor_dim1[15:0]=16
    g1[3] = (int)(512u << 16);  // tensor_dim1 hi=0 | tile_dim0=512
    g1[4] = 16;                 // tile_dim1=16 | tile_dim2=0
    g1[5] = 512;                // tensor_dim0_stride low32 = 512
    g1[6] = 0;                  // stride0 hi | stride1 lo
    g1[7] = 0;                  // stride1 hi
    v4i gz; gz[0] = 0; gz[1] = 0; gz[2] = 0; gz[3] = 0;
#if defined(__clang_major__) && (__clang_major__ >= 23)
    v8i gz8; gz8[0]=0; gz8[1]=0; gz8[2]=0; gz8[3]=0; gz8[4]=0; gz8[5]=0; gz8[6]=0; gz8[7]=0;
    __builtin_amdgcn_tensor_load_to_lds(g0, g1, gz, gz, gz8, 0);
#else
    __builtin_amdgcn_tensor_load_to_lds(g0, g1, gz, gz, 0);
#endif
    __builtin_amdgcn_s_wait_tensorcnt((short)0);
  }
  __syncthreads();

  const int mt = mtg * 4 + wave;                   // 0..255
  const float* A = inp + mt * 16 * 512;
  v8f acc = {};
  for (int k = 0; k < 512; k += 4) {
    const int ka = k + 2 * hf;
    v2f a; a.x = A[r * 512 + ka];        a.y = A[r * 512 + ka + 1];
    v2f b; b.x = ldsB[r * 512 + ka];     b.y = ldsB[r * 512 + ka + 1];
    acc = __builtin_amdgcn_wmma_f32_16x16x4_f32(false, a, false, b,
                                                (short)0, acc, false, false);
  }
  const int jcol = nt * 16 + r;
  const float bb = bih[jcol] + bhh[jcol];
  float* D = pre + (mt * 16) * 512 + nt * 16;
#pragma unroll
  for (int v = 0; v < 8; ++v) D[(v + 8 * hf) * 512 + r] = acc[v] + bb;
}

// 13) sequential LSTM recurrence: one block per direction, thread j owns
//     gate row j with its Whh row pinned in 128 VGPRs; h/c/gates in LDS.
//     Next timestep's preactivations are prefetched (global_prefetch_b8).
__global__ __launch_bounds__(512) void k_lstm(const float* __restrict__ preF,
                                              const float* __restrict__ preB,
                                              const float* __restrict__ WhhF,
                                              const float* __restrict__ WhhB,
                                              float* __restrict__ out1) {
  const int dir = blockIdx.x;
  const float* pre = dir ? preB : preF;
  const float* Whh = dir ? WhhB : WhhF;
  const int j = threadIdx.x;  // gate row 0..511
  __shared__ float sh_h[8][128];
  __shared__ float sh_c[8][128];
  __shared__ float sh_g[8][512];
  float w[128];
#pragma unroll
  for (int k = 0; k < 128; ++k) w[k] = Whh[j * 128 + k];
  for (int p = 0; p < 2; ++p) {
    const int id = j + p * 512;
    sh_h[id >> 7][id & 127] = 0.f;
    sh_c[id >> 7][id & 127] = 0.f;
  }
  __syncthreads();
  for (int s = 0; s < 512; ++s) {
    const int t = dir ? (511 - s) : s;
    if (s + 1 < 512) {
      const int t2 = dir ? (510 - s) : (s + 1);
#pragma unroll
      for (int n = 0; n < 8; ++n)
        __builtin_prefetch(&pre[(n * 512 + t2) * 512 + j], 0, 1);
    }
    for (int n = 0; n < 8; ++n) {
      float acc = pre[(n * 512 + t) * 512 + j];
      const float4* hp = (const float4*)(&sh_h[n][0]);
#pragma unroll
      for (int k4 = 0; k4 < 32; ++k4) {
        const float4 hv = hp[k4];
        acc += w[4 * k4 + 0] * hv.x + w[4 * k4 + 1] * hv.y +
               w[4 * k4 + 2] * hv.z + w[4 * k4 + 3] * hv.w;
      }
      sh_g[n][j] = acc;
    }
    __syncthreads();
    for (int p = 0; p < 2; ++p) {
      const int id = j + p * 512;
      const int n = id >> 7, k = id & 127;
      const float ig = sh_g[n][k],       fg = sh_g[n][128 + k];
      const float gg = sh_g[n][256 + k], og = sh_g[n][384 + k];
      const float c = sigf(fg) * sh_c[n][k] + sigf(ig) * tanhf(gg);
      const float h = sigf(og) * tanhf(c);
      sh_c[n][k] = c;
      sh_h[n][k] = h;
      out1[(n * 512 + t) * 256 + dir * 128 + k] = h;
    }
    __syncthreads();
  }
}

// ---------------------------------------------------------------------------
extern "C" void kernel_launch(void* const* d_in, const int* in_sizes, int n_in,
                              void* d_out, int out_size, void* d_ws, size_t ws_size,
                              hipStream_t stream) {
  (void)in_sizes; (void)n_in; (void)out_size;
  const float* doc   = (const float*)d_in[0];
  const float* qv    = (const float*)d_in[1];
  // d_in[2] context_mask: unused by reference math
  const float* emap  = (const float*)d_in[3];
  const float* emask = (const float*)d_in[4];
  const float* qw    = (const float*)d_in[5];
  const float* Wt    = (const float*)d_in[6];
  const float* at    = (const float*)d_in[7];
  const float* W1    = (const float*)d_in[8];
  const float* W2    = (const float*)d_in[9];
  const float* WihF  = (const float*)d_in[10];
  const float* WhhF  = (const float*)d_in[11];
  const float* bihF  = (const float*)d_in[12];
  const float* bhhF  = (const float*)d_in[13];
  const float* WihB  = (const float*)d_in[14];
  const float* WhhB  = (const float*)d_in[15];
  const float* bihB  = (const float*)d_in[16];
  const float* bhhB  = (const float*)d_in[17];
  // d_in[18] context_lens: unused
  const int*   adj   = (const int*)d_in[19];

  float* out1 = (float*)d_out;              // doc_out      (8,512,256)
  float* out2 = out1 + 8 * 512 * 256;       // entity_state2(8,128,256)
  float* out3 = out2 + 8 * 128 * 256;       // softmask     (8,128,1)

  // workspace carve-up (floats); total ~8.44M floats = ~34 MB
  float* ws    = (float*)d_ws;
  float* ent   = ws;                    // 1024*512
  float* query = ent   + 524288;        // 4096
  float* adjm  = query + 4096;          // 1024
  float* hall  = adjm  + 1024;          // 12*1024*64
  float* qg1   = hall  + 786432;        // 12288
  float* qg2   = qg1   + 12288;         // 12288
  float* src   = qg2   + 12288;         // 12288
  float* dst   = src   + 12288;         // 12288
  float* hlast = dst   + 12288;         // 4*8*128*64
  float* coefs = hlast + 262144;        // 4*8*128*128
  float* inp   = coefs + 524288;        // 8*512*512
  float* preF  = inp   + 2097152;       // 4096*512
  float* preB  = preF  + 2097152;       // 4096*512
  (void)ws_size;

  k_entity_pool<<<1024, 256, 0, stream>>>(doc, emap, ent);
  k_query_proj <<<16,   256, 0, stream>>>(qv, qw, query);
  k_softmask   <<<4,    256, 0, stream>>>(ent, query, emask, out3, adjm);
  k_hall       <<<768,  128, 0, stream>>>(ent, Wt, hall);        // WMMA
  k_qg1        <<<48,   256, 0, stream>>>(qv, W1, qg1);
  k_qg2        <<<48,   256, 0, stream>>>(qg1, W2, qg2);
  k_srcdst     <<<48,   256, 0, stream>>>(hall, qg2, at, src, dst);
  k_hlast      <<<1024, 256, 0, stream>>>(hall, adjm, hlast);
  k_coefs      <<<4096, 128, 0, stream>>>(src, dst, adj, coefs);
  k_heads      <<<256,  128, 0, stream>>>(coefs, hlast, out2);   // WMMA
  k_expand     <<<1024, 128, 0, stream>>>(emap, out2, inp);      // WMMA
  k_copy_doc   <<<4096, 256, 0, stream>>>(doc, inp);
  k_pre        <<<2048, 128, 0, stream>>>(inp, WihF, bihF, bhhF, preF); // WMMA+TDM
  k_pre        <<<2048, 128, 0, stream>>>(inp, WihB, bihB, bhhB, preB); // WMMA+TDM
  k_lstm       <<<2,    512, 0, stream>>>(preF, preB, WhhF, WhhB, out1);
}